// TopK_80058190397641
// MI455X (gfx1250) — compile-verified
//
#include <hip/hip_runtime.h>
#include <stdint.h>

// ---------------- problem constants ----------------
#define ROWLEN   32768          // elements per row
#define NT       1024           // threads per block (32 wave32s)
#define EPT      (ROWLEN / NT)  // 32 elements per thread
#define V4PT     (ROWLEN / 4 / NT) // 8 float4 chunks per thread
#define NBINS    1024           // 10-bit radix digit
#define DIGBITS  10
#define CAND_MAX 1024
#define KSEL     32

// ---------------- CDNA5 async global<->LDS DMA ----------------
#if __has_builtin(__builtin_amdgcn_global_load_async_to_lds_b128) && \
    __has_builtin(__builtin_amdgcn_global_store_async_from_lds_b128)
#define USE_ASYNC_LDS 1
#else
#define USE_ASYNC_LDS 0
#endif

typedef int v4i __attribute__((ext_vector_type(4)));
typedef __attribute__((address_space(3))) char lds_char;
typedef __attribute__((address_space(1))) char glb_char;
typedef __attribute__((address_space(1))) v4i  glb_v4i;   // global int4
typedef __attribute__((address_space(3))) v4i  lds_v4i;   // LDS int4

__device__ __forceinline__ void wait_async0() {
#if __has_builtin(__builtin_amdgcn_s_wait_asynccnt)
  __builtin_amdgcn_s_wait_asynccnt(0);
#else
  asm volatile("s_wait_asynccnt 0" ::: "memory");
#endif
}

// Monotonic composite sort key: 32-bit order-preserving float key in the high
// bits, (32767 - idx) in the low 15 bits. Strictly larger composite key ==
// "earlier in jax.lax.top_k order" (bigger value, or equal value w/ lower idx).
__device__ __forceinline__ uint64_t make_ckey(float f, int idx) {
  uint32_t u = __float_as_uint(f);
  uint32_t k = (u & 0x80000000u) ? ~u : (u | 0x80000000u);
  return ((uint64_t)k << 15) | (uint64_t)(32767 - idx);
}

__global__ __launch_bounds__(NT)
void topk_relu_scatter_kernel(const float* __restrict__ x, float* __restrict__ out) {
  __shared__ __align__(16) float s_row[ROWLEN];   // 128 KB row cache
  __shared__ uint32_t s_hist[NBINS];              // 4 KB radix histogram
  __shared__ uint64_t s_cand[CAND_MAX];           // 8 KB candidate keys
  __shared__ uint32_t s_candCnt;
  __shared__ uint64_t s_hi;                       // resolved key prefix
  __shared__ uint64_t s_kth;                      // K-th largest composite key
  __shared__ int s_shift, s_remK, s_M;

  const int tid = threadIdx.x;
  const size_t rowBase = (size_t)blockIdx.x * (size_t)ROWLEN;

  // ---- Phase 1: DMA row into LDS (async path, tracked by ASYNCcnt) ----
#if USE_ASYNC_LDS
  {
    glb_char* g = (glb_char*)(x + rowBase);
    lds_char* l = (lds_char*)s_row;
    for (int c = 0; c < V4PT; ++c) {
      const int v4 = c * NT + tid;   // coalesced 16B chunks across the wave
      __builtin_amdgcn_global_load_async_to_lds_b128(
          (glb_v4i*)(g + (size_t)v4 * 16),
          (lds_v4i*)(l + (size_t)v4 * 16),
          /*imm offset*/0, /*cpol*/0);
    }
    wait_async0();   // this wave's ASYNCcnt -> 0; barrier below publishes LDS
  }
#else
  {
    const float4* g = (const float4*)(x + rowBase);
    float4* l = (float4*)s_row;
    for (int c = 0; c < V4PT; ++c) { const int v4 = c * NT + tid; l[v4] = g[v4]; }
  }
#endif
  __syncthreads();

  // ---- Phase 2: radix-select the K-th largest composite key ----
  if (tid == 0) {
    s_hi = 0; s_shift = 47; s_remK = KSEL; s_M = ROWLEN; s_candCnt = 0;
  }
  __syncthreads();

  for (;;) {
    const int      M     = s_M;
    const int      shift = s_shift;
    const uint64_t hi    = s_hi;
    const int      remK  = s_remK;
    if (M <= CAND_MAX) break;    // uniform: state read after a barrier

    const int      d  = (shift < DIGBITS) ? shift : DIGBITS;
    const int      ns = shift - d;
    const uint32_t dm = (1u << d) - 1u;

    for (int b = tid; b < NBINS; b += NT) s_hist[b] = 0;
    __syncthreads();

    for (int e = 0; e < EPT; ++e) {
      const int i = e * NT + tid;                  // bank-conflict-free stride
      const uint64_t ck = make_ckey(s_row[i], i);
      if ((ck >> shift) == hi)
        atomicAdd(&s_hist[(uint32_t)(ck >> ns) & dm], 1u);
    }
    __syncthreads();

    if (tid == 0) {  // suffix scan from the top bin; typically ~250 iters, once
      uint32_t acc = 0;
      int g = 0;
      for (int b = (int)dm; b >= 0; --b) {
        const uint32_t h = s_hist[b];
        if (acc + h >= (uint32_t)remK) {
          g = b; s_remK = remK - (int)acc; s_M = (int)h;
          break;
        }
        acc += h;
      }
      s_hi    = (hi << d) | (uint64_t)(uint32_t)g;
      s_shift = ns;
    }
    __syncthreads();
  }

  // ---- Phase 3: gather candidates matching the resolved prefix ----
  {
    const int      shift = s_shift;
    const uint64_t hi    = s_hi;
    for (int e = 0; e < EPT; ++e) {
      const int i = e * NT + tid;
      const uint64_t ck = make_ckey(s_row[i], i);
      if ((ck >> shift) == hi) {
        const uint32_t p = atomicAdd(&s_candCnt, 1u);
        if (p < CAND_MAX) s_cand[p] = ck;
      }
    }
  }
  __syncthreads();

  // ---- Phase 4: exact K-th key among <=1024 candidates (keys distinct) ----
  {
    const int M    = (int)s_candCnt;
    const int remK = s_remK;
    if (tid < M) {
      const uint64_t c = s_cand[tid];
      int cnt = 0;
      for (int j = 0; j < M; ++j) cnt += (s_cand[j] > c) ? 1 : 0;   // broadcast reads
      if (cnt == remK - 1) s_kth = c;                               // exactly one writer
    }
  }
  __syncthreads();
  const uint64_t kth = s_kth;

  // ---- Phase 5: transform in LDS, then DMA row back out ----
  for (int e = 0; e < EPT; ++e) {
    const int i = e * NT + tid;
    const float v = s_row[i];
    const uint64_t ck = make_ckey(v, i);
    s_row[i] = (ck >= kth) ? fmaxf(v, 0.0f) : 0.0f;
  }
  __syncthreads();

#if USE_ASYNC_LDS
  {
    glb_char* g = (glb_char*)(out + rowBase);
    lds_char* l = (lds_char*)s_row;
    for (int c = 0; c < V4PT; ++c) {
      const int v4 = c * NT + tid;
      __builtin_amdgcn_global_store_async_from_lds_b128(
          (glb_v4i*)(g + (size_t)v4 * 16),
          (lds_v4i*)(l + (size_t)v4 * 16),
          /*imm offset*/0, /*cpol*/0);
    }
    wait_async0();   // stores complete before ENDPGM
  }
#else
  {
    float4* g = (float4*)(out + rowBase);
    const float4* l = (const float4*)s_row;
    for (int c = 0; c < V4PT; ++c) { const int v4 = c * NT + tid; g[v4] = l[v4]; }
  }
#endif
}

extern "C" void kernel_launch(void* const* d_in, const int* in_sizes, int n_in,
                              void* d_out, int out_size, void* d_ws, size_t ws_size,
                              hipStream_t stream) {
  (void)n_in; (void)out_size; (void)d_ws; (void)ws_size;
  const float* x = (const float*)d_in[0];
  float* out = (float*)d_out;
  const int rows = in_sizes[0] / ROWLEN;   // 4096 for the reference shapes
  if (rows <= 0) return;
  topk_relu_scatter_kernel<<<dim3((unsigned)rows), dim3(NT), 0, stream>>>(x, out);
}